// GATLayer_3341484556623
// MI455X (gfx1250) — compile-verified
//
#include <hip/hip_runtime.h>
#include <hip/hip_bf16.h>
#include <math.h>

// GAT layer: N=50000 nodes, E=800000 edges, FIN=128, NH=4 heads, FOUT=16.
#define NN   50000
#define NE   800000
#define FIN  128
#define NH   4
#define FOUT 16
#define CC   64   // NH*FOUT

typedef __attribute__((ext_vector_type(2))) float v2f;
typedef __attribute__((ext_vector_type(8))) float v8f;

// ---------------------------------------------------------------------------
// K0: zero accumulators (agg [N,64], denom [N,4], gmax scalar).
// ---------------------------------------------------------------------------
__global__ __launch_bounds__(256) void init_ws(float* __restrict__ agg,
                                               float* __restrict__ denom,
                                               unsigned* __restrict__ gmax) {
    int idx = blockIdx.x * 256 + threadIdx.x;
    if (idx < NN * CC) agg[idx] = 0.0f;
    if (idx < NN * NH) denom[idx] = 0.0f;
    if (idx == 0) *gmax = 0u;   // ordered-uint: 0 is below every real score
}

// ---------------------------------------------------------------------------
// K1: dual GEMM with V_WMMA_F32_16X16X4_F32.
//   proj = x @ w_proj^T, skip = x @ w_skip^T   (both [N, 64])
// One wave per 16x16 output tile; block = 8 waves = 4 col-tiles x {proj,skip}.
// A frag (16x4 of x) and B frag (4x16 = w rows are output cols, K-contiguous)
// are each a single aligned float2 per lane:
//   lanes 0-15:  K = {k, k+1};  lanes 16-31: K = {k+2, k+3}
// C/D layout: VGPR j holds M = j + 8*(lane>=16), N = lane&15.
// ---------------------------------------------------------------------------
__global__ __launch_bounds__(256) void gemm_proj_skip(
    const float* __restrict__ x, const float* __restrict__ w_proj,
    const float* __restrict__ w_skip, float* __restrict__ proj,
    float* __restrict__ skipb) {
    const int wave = threadIdx.x >> 5;
    const int lane = threadIdx.x & 31;
    const int half = lane >> 4;        // 0: K={k,k+1}, 1: K={k+2,k+3}
    const int l16  = lane & 15;
    const int rowBase = blockIdx.x * 16;        // 50000/16 = 3125 exact
    const int colBase = (wave & 3) * 16;
    const float* W   = (wave < 4) ? w_proj : w_skip;
    float*       out = (wave < 4) ? proj   : skipb;

    const float* aptr = x + (size_t)(rowBase + l16) * FIN + 2 * half;
    const float* bptr = W + (size_t)(colBase + l16) * FIN + 2 * half;

    v8f acc = {};
#pragma unroll 4
    for (int k = 0; k < FIN; k += 4) {
        v2f a = *(const v2f*)(aptr + k);   // 8B-aligned
        v2f b = *(const v2f*)(bptr + k);
        acc = __builtin_amdgcn_wmma_f32_16x16x4_f32(
            /*neg_a=*/false, a, /*neg_b=*/false, b,
            /*c_mod=*/(short)0, acc, /*reuse_a=*/false, /*reuse_b=*/false);
    }
#pragma unroll
    for (int j = 0; j < 8; ++j) {
        int m = j + 8 * half;
        out[(size_t)(rowBase + m) * CC + colBase + l16] = acc[j];
    }
}

// ---------------------------------------------------------------------------
// K2: per-node attention logits  s_src/s_trg [N, NH]
//   s_src[n,h] = sum_f proj[n,h,f] * a_src[h,f]
// idx = n*NH + h  ->  proj offset idx*FOUT == n*64 + h*16 (contiguous 64B).
// ---------------------------------------------------------------------------
__global__ __launch_bounds__(256) void node_scores(
    const float* __restrict__ proj, const float* __restrict__ a_src,
    const float* __restrict__ a_trg, float* __restrict__ s_src,
    float* __restrict__ s_trg) {
    int idx = blockIdx.x * 256 + threadIdx.x;
    if (idx >= NN * NH) return;
    int h = idx & (NH - 1);
    const float* p = proj + (size_t)idx * FOUT;
    float s1 = 0.0f, s2 = 0.0f;
#pragma unroll
    for (int f = 0; f < FOUT; ++f) {
        float v = p[f];
        s1 += v * a_src[h * FOUT + f];
        s2 += v * a_trg[h * FOUT + f];
    }
    s_src[idx] = s1;
    s_trg[idx] = s2;
}

// ---------------------------------------------------------------------------
// K3: edge scores = leaky_relu(s_src[src]+s_trg[trg], 0.2) and global max
// via order-preserving float->uint map + one atomicMax per block.
// ---------------------------------------------------------------------------
__global__ __launch_bounds__(256) void edge_scores(
    const long long* __restrict__ src, const long long* __restrict__ trg,
    const float* __restrict__ s_src, const float* __restrict__ s_trg,
    float* __restrict__ scores, unsigned* __restrict__ gmax) {
    __shared__ float smax[256];
    int idx = blockIdx.x * 256 + threadIdx.x;
    float sc = -3.4e38f;
    if (idx < NE * NH) {
        int e = idx >> 2;           // NH == 4
        int h = idx & 3;
        int s = (int)src[e];
        int t = (int)trg[e];
        float v = s_src[s * NH + h] + s_trg[t * NH + h];
        sc = (v > 0.0f) ? v : 0.2f * v;
        scores[idx] = sc;
    }
    smax[threadIdx.x] = sc;
    __syncthreads();
#pragma unroll
    for (int off = 128; off > 0; off >>= 1) {
        if (threadIdx.x < off)
            smax[threadIdx.x] = fmaxf(smax[threadIdx.x], smax[threadIdx.x + off]);
        __syncthreads();
    }
    if (threadIdx.x == 0) {
        unsigned u = __float_as_uint(smax[0]);
        u = (u & 0x80000000u) ? ~u : (u | 0x80000000u);
        atomicMax(gmax, u);
    }
}

// ---------------------------------------------------------------------------
// K4: exp_scores = exp(score - gmax) (in place) and segment-sum denominator
// over target nodes via global_atomic_add_f32.
// ---------------------------------------------------------------------------
__global__ __launch_bounds__(256) void exp_denom(
    const long long* __restrict__ trg, float* __restrict__ scores,
    const unsigned* __restrict__ gmax, float* __restrict__ denom) {
    int idx = blockIdx.x * 256 + threadIdx.x;
    if (idx >= NE * NH) return;
    unsigned u = *gmax;
    float m = (u & 0x80000000u) ? __uint_as_float(u & 0x7FFFFFFFu)
                                : __uint_as_float(~u);
    int e = idx >> 2;
    int h = idx & 3;
    float es = expf(scores[idx] - m);
    scores[idx] = es;
    atomicAdd(&denom[(int)trg[e] * NH + h], es);
}

// ---------------------------------------------------------------------------
// K5: attention-weighted aggregation (the bandwidth-dominant phase):
//   agg[trg, c] += proj[src, c] * exp_scores[e, h] / (denom[trg, h] + 1e-16)
// One thread per (edge, channel): 51.2M threads, coalesced proj gather,
// float atomic scatter-add.
// ---------------------------------------------------------------------------
__global__ __launch_bounds__(256) void aggregate(
    const long long* __restrict__ src, const long long* __restrict__ trg,
    const float* __restrict__ proj, const float* __restrict__ scores,
    const float* __restrict__ denom, float* __restrict__ agg) {
    long long idx = (long long)blockIdx.x * 256 + threadIdx.x;
    if (idx >= (long long)NE * CC) return;
    int e = (int)(idx >> 6);
    int c = (int)(idx & 63);
    int h = c >> 4;
    int t = (int)trg[e];
    int s = (int)src[e];
    float attn = scores[e * NH + h] / (denom[t * NH + h] + 1e-16f);
    atomicAdd(&agg[(size_t)t * CC + c], proj[(size_t)s * CC + c] * attn);
}

// ---------------------------------------------------------------------------
// K6: out = elu(agg + skip)
// ---------------------------------------------------------------------------
__global__ __launch_bounds__(256) void finalize(
    const float* __restrict__ agg, const float* __restrict__ skipb,
    float* __restrict__ out) {
    int idx = blockIdx.x * 256 + threadIdx.x;
    if (idx >= NN * CC) return;
    float v = agg[idx] + skipb[idx];
    out[idx] = (v > 0.0f) ? v : (expf(v) - 1.0f);
}

// ---------------------------------------------------------------------------
extern "C" void kernel_launch(void* const* d_in, const int* in_sizes, int n_in,
                              void* d_out, int out_size, void* d_ws, size_t ws_size,
                              hipStream_t stream) {
    const float*     x      = (const float*)d_in[0];
    const long long* eidx   = (const long long*)d_in[1];  // int64 [2, E]
    const float*     w_proj = (const float*)d_in[2];      // [64, 128]
    const float*     a_src  = (const float*)d_in[3];      // [4, 16]
    const float*     a_trg  = (const float*)d_in[4];      // [4, 16]
    const float*     w_skip = (const float*)d_in[5];      // [64, 128]
    float*           out    = (float*)d_out;              // [N, 64] f32

    const long long* src = eidx;          // row 0
    const long long* trg = eidx + NE;     // row 1

    // Workspace layout (floats); total ~13.4M floats (~53.6 MB).
    float* ws      = (float*)d_ws;
    float* proj    = ws;                       // N*64
    float* skipb   = proj  + (size_t)NN * CC;  // N*64
    float* s_src   = skipb + (size_t)NN * CC;  // N*4
    float* s_trg   = s_src + (size_t)NN * NH;  // N*4
    float* scores  = s_trg + (size_t)NN * NH;  // E*4 (scores, then exp_scores)
    float* denom   = scores + (size_t)NE * NH; // N*4
    float* agg     = denom + (size_t)NN * NH;  // N*64
    unsigned* gmax = (unsigned*)(agg + (size_t)NN * CC);  // 1

    const int ncells   = NN * CC;              // 3.2M
    const int nescores = NE * NH;              // 3.2M

    init_ws<<<(ncells + 255) / 256, 256, 0, stream>>>(agg, denom, gmax);

    gemm_proj_skip<<<NN / 16, 256, 0, stream>>>(x, w_proj, w_skip, proj, skipb);

    node_scores<<<(NN * NH + 255) / 256, 256, 0, stream>>>(proj, a_src, a_trg,
                                                           s_src, s_trg);

    edge_scores<<<(nescores + 255) / 256, 256, 0, stream>>>(src, trg, s_src,
                                                            s_trg, scores, gmax);

    exp_denom<<<(nescores + 255) / 256, 256, 0, stream>>>(trg, scores, gmax,
                                                          denom);

    long long nagg = (long long)NE * CC;       // 51.2M
    aggregate<<<(unsigned)((nagg + 255) / 256), 256, 0, stream>>>(
        src, trg, proj, scores, denom, agg);

    finalize<<<(ncells + 255) / 256, 256, 0, stream>>>(agg, skipb, out);
}